// SEP_2044404433512
// MI455X (gfx1250) — compile-verified
//
#include <hip/hip_runtime.h>
#include <math.h>

#define THW_N   1568
#define B_N     16
#define C_N     1024
#define S_N     128
#define KC      128     // K chunk staged in LDS
#define APAD    130     // padded LDS row stride (floats), even for b64 alignment

typedef float v8f __attribute__((ext_vector_type(8)));
typedef float v2f __attribute__((ext_vector_type(2)));

// ---------------- pooling reductions ----------------
// video: (THW, B, C) -> ctx[b*C+c] = mean_thw + max_thw
__global__ __launch_bounds__(256) void spatial_red(const float* __restrict__ video,
                                                   float* __restrict__ ctx) {
    int j = blockIdx.x * 256 + threadIdx.x;          // j = b*C + c, 16384 total
    const float* p = video + j;                      // (thw*B + b)*C + c = thw*16384 + j
    float sum = 0.f, mx = -INFINITY;
    for (int t = 0; t < THW_N; ++t) {
        float v = p[(size_t)t * (B_N * C_N)];
        sum += v; mx = fmaxf(mx, v);
    }
    ctx[j] = sum * (1.0f / (float)THW_N) + mx;
}

// text: (B, S, C) -> ctx[b*C+c] = mean_s + max_s
__global__ __launch_bounds__(256) void txt_red(const float* __restrict__ text,
                                               float* __restrict__ ctx) {
    int j = blockIdx.x * 256 + threadIdx.x;          // b*C + c
    int b = j >> 10, c = j & (C_N - 1);
    const float* p = text + (size_t)b * S_N * C_N + c;
    float sum = 0.f, mx = -INFINITY;
    for (int s = 0; s < S_N; ++s) {
        float v = p[(size_t)s * C_N];
        sum += v; mx = fmaxf(mx, v);
    }
    ctx[j] = sum * (1.0f / (float)S_N) + mx;
}

// ---------------- fused attn-build + batched GEMM (f32 WMMA) ----------------
// grid.x = B_N * 16 ; block = 256 (8 waves). Each block: batch b, 64 output rows.
// Output written as (b, s, c) into ws for contiguous LayerNorm rows.
__global__ __launch_bounds__(256) void fused_gemm(
    const float* __restrict__ text,     // (B,S,C)
    const float* __restrict__ weight,   // (C,C)
    const float* __restrict__ conv_w,   // 9
    const float* __restrict__ conv_b,   // 1
    const float* __restrict__ bn_gamma, const float* __restrict__ bn_beta,
    const float* __restrict__ bn_mean,  const float* __restrict__ bn_var,
    const float* __restrict__ spatial,  // (B,C) ws
    const float* __restrict__ txtc,     // (B,C) ws
    float* __restrict__ out)            // (B,S,C) ws
{
    __shared__ float A[64 * APAD];      // fused attn strip chunk
    __shared__ float xs[C_N];           // txt ctx for this batch
    __shared__ float ps[66], qs[66];    // per-row BN-rank1 constants (with halo)
    __shared__ float cw[9];
    __shared__ float cbias;

    const int b      = blockIdx.x >> 4;
    const int o_base = (blockIdx.x & 15) * 64;
    const int tid    = threadIdx.x;

    for (int i = tid; i < C_N; i += 256) xs[i] = txtc[b * C_N + i];
    if (tid < 66) {
        int r = o_base - 1 + tid;
        float p = 0.f, q = 0.f;                      // p=q=0 => relu(...)==0 (zero pad rows)
        if (r >= 0 && r < C_N) {
            float inv = bn_gamma[r] * rsqrtf(bn_var[r] + 1e-5f);
            p = spatial[b * C_N + r] * inv;
            q = bn_beta[r] - bn_mean[r] * inv;
        }
        ps[tid] = p; qs[tid] = q;
    }
    if (tid < 9)  cw[tid] = conv_w[tid];
    if (tid == 0) cbias = conv_b[0];

    const int wave  = tid >> 5;
    const int lane  = tid & 31;
    const int msub  = wave & 3;          // which 16-row subtile of the 64
    const int nhalf = wave >> 2;         // s columns 0-63 or 64-127
    const int lo    = lane & 15;
    const int hi    = lane >> 4;

    v8f acc[4] = {};                     // 4 N-tiles of 16x16 f32

    const float* trow = text + (size_t)b * S_N * C_N;

    for (int k0 = 0; k0 < C_N; k0 += KC) {
        __syncthreads();                 // A reuse barrier (also covers init loads)
        // ---- phase 1: build fused A chunk: sigmoid(conv3x3(relu(p*x+q)))*weight
        for (int e = tid; e < 64 * KC; e += 256) {
            int ol = e >> 7;             // KC == 128
            int il = e & (KC - 1);
            int i  = k0 + il;
            float a = cbias;
            #pragma unroll
            for (int dy = 0; dy < 3; ++dy) {
                float p = ps[ol + dy];
                float q = qs[ol + dy];
                #pragma unroll
                for (int dx = 0; dx < 3; ++dx) {
                    int j = i + dx - 1;
                    float g = 0.f;
                    if (j >= 0 && j < C_N) {
                        g = p * xs[j] + q;
                        g = g > 0.f ? g : 0.f;
                    }
                    a += cw[dy * 3 + dx] * g;
                }
            }
            float sig = 1.0f / (1.0f + __expf(-a));
            A[ol * APAD + il] = sig * weight[(size_t)(o_base + ol) * C_N + i];
        }
        __syncthreads();
        // ---- phase 2: WMMA over the chunk, K step 4
        for (int k = 0; k < KC; k += 4) {
            // A fragment: lane holds A[M=lo][K = 2*hi + v], v in {0,1}
            v2f af = *(const v2f*)&A[(msub * 16 + lo) * APAD + (k + hi * 2)];
            int gi = k0 + k + hi * 2;
            #pragma unroll
            for (int nt = 0; nt < 4; ++nt) {
                int s = nhalf * 64 + nt * 16 + lo;
                // B fragment: B[K][N] = text[b, s=N, i=K]; contiguous K per lane
                v2f bf = *(const v2f*)&trow[(size_t)s * C_N + gi];
                acc[nt] = __builtin_amdgcn_wmma_f32_16x16x4_f32(
                    false, af, false, bf, (short)0, acc[nt], false, false);
            }
        }
    }

    // ---- store: D lane layout: VGPR v, lane -> (M = hi*8 + v, N = lo)
    #pragma unroll
    for (int nt = 0; nt < 4; ++nt) {
        int s = nhalf * 64 + nt * 16 + lo;
        int c = o_base + msub * 16 + hi * 8;
        float* dst = out + ((size_t)(b * S_N + s)) * C_N + c;
        *(float4*)(dst)     = *((float4*)&acc[nt]);
        *(float4*)(dst + 4) = *((float4*)&acc[nt] + 1);
    }
}

// ---------------- bias + LayerNorm -> (S, B, C) ----------------
__global__ __launch_bounds__(256) void ln_kernel(
    const float* __restrict__ gout,     // (B,S,C) ws
    const float* __restrict__ bias,
    const float* __restrict__ lgam, const float* __restrict__ lbet,
    float* __restrict__ out)            // (S,B,C)
{
    int row = blockIdx.x;               // row = s*B + b  (matches output layout)
    int s = row >> 4, b = row & 15;
    int tid = threadIdx.x;
    const float* src = gout + ((size_t)(b * S_N + s)) * C_N;

    float4 x  = *(const float4*)&src[tid * 4];
    float4 bb = *(const float4*)&bias[tid * 4];
    x.x += bb.x; x.y += bb.y; x.z += bb.z; x.w += bb.w;

    __shared__ float rs[256], rq[256];
    rs[tid] = x.x + x.y + x.z + x.w;
    rq[tid] = x.x * x.x + x.y * x.y + x.z * x.z + x.w * x.w;
    __syncthreads();
    for (int off = 128; off > 0; off >>= 1) {
        if (tid < off) { rs[tid] += rs[tid + off]; rq[tid] += rq[tid + off]; }
        __syncthreads();
    }
    float mu  = rs[0] * (1.0f / (float)C_N);
    float var = rq[0] * (1.0f / (float)C_N) - mu * mu;
    float inv = rsqrtf(var + 1e-12f);

    float4 g  = *(const float4*)&lgam[tid * 4];
    float4 be = *(const float4*)&lbet[tid * 4];
    float4 o;
    o.x = (x.x - mu) * inv * g.x + be.x;
    o.y = (x.y - mu) * inv * g.y + be.y;
    o.z = (x.z - mu) * inv * g.z + be.z;
    o.w = (x.w - mu) * inv * g.w + be.w;
    *(float4*)&out[(size_t)row * C_N + tid * 4] = o;
}

extern "C" void kernel_launch(void* const* d_in, const int* in_sizes, int n_in,
                              void* d_out, int out_size, void* d_ws, size_t ws_size,
                              hipStream_t stream) {
    const float* video    = (const float*)d_in[0];
    const float* text     = (const float*)d_in[1];
    const float* weight   = (const float*)d_in[2];
    const float* bias     = (const float*)d_in[3];
    const float* conv_w   = (const float*)d_in[4];
    const float* conv_b   = (const float*)d_in[5];
    const float* bn_gamma = (const float*)d_in[6];
    const float* bn_beta  = (const float*)d_in[7];
    const float* bn_mean  = (const float*)d_in[8];
    const float* bn_var   = (const float*)d_in[9];
    const float* ln_gamma = (const float*)d_in[10];
    const float* ln_beta  = (const float*)d_in[11];
    float* outp = (float*)d_out;

    float* wsf     = (float*)d_ws;
    float* spatial = wsf;                        // 16384 floats
    float* txtc    = wsf + B_N * C_N;            // 16384 floats
    float* gemmbuf = wsf + 2 * B_N * C_N;        // B*S*C = 2,097,152 floats

    spatial_red<<<(B_N * C_N) / 256, 256, 0, stream>>>(video, spatial);
    txt_red<<<(B_N * C_N) / 256, 256, 0, stream>>>(text, txtc);
    fused_gemm<<<B_N * 16, 256, 0, stream>>>(text, weight, conv_w, conv_b,
                                             bn_gamma, bn_beta, bn_mean, bn_var,
                                             spatial, txtc, gemmbuf);
    ln_kernel<<<S_N * B_N, 256, 0, stream>>>(gemmbuf, bias, ln_gamma, ln_beta, outp);
}